// RoiPooling_84705345012203
// MI455X (gfx1250) — compile-verified
//
#include <hip/hip_runtime.h>
#include <hip/hip_bf16.h>

// -------- gfx1250 async global->LDS copy path (inline asm; see CDNA5 ISA) -----
// GV addressing form: "global_load_async_to_lds_b128 vdst(LDS byte addr), vaddr(64b), off"
// Tracked by ASYNCcnt; loads complete in order -> constant-immediate s_wait_asynccnt
// gives deterministic pipelining. Flat-shared low 32 bits == LDS byte offset (ISA 10.2).
__device__ __forceinline__ void async_copy16(const float* g, float* l) {
  unsigned lo = (unsigned)(unsigned long long)l;  // LDS byte offset
  asm volatile("global_load_async_to_lds_b128 %0, %1, off"
               :: "v"(lo), "v"(g) : "memory");
}

template <int N>
__device__ __forceinline__ void wait_async() {
  asm volatile("s_wait_asynccnt %0" :: "i"(N) : "memory");
}

// Problem constants: img [4,32,32,256] f32, rois [4,64,4], out [4*64, 21*256].
#define IMG_H 32
#define IMG_W 32
#define IMG_C 256
#define N_ROI 64
#define MAXW  24   // staged columns per row segment (actual width <= 20 by setup)

__global__ __launch_bounds__(128) void roi_pool_gfx1250(
    const float* __restrict__ img, const float* __restrict__ rois,
    float* __restrict__ out)
{
  // [wave][slot][col*32 + lane]  -> 4 * 4 * 768 * 4B = 48 KB
  __shared__ float buf[4][4][MAXW * 32];

  const int tid  = threadIdx.x;
  const int lane = tid & 31;
  const int wave = tid >> 5;

  const int bid    = blockIdx.x;   // 0..511
  const int split  = bid & 1;      // which 128-channel half
  const int roiId  = bid >> 1;     // 0..255 == b*64 + ri
  const int b      = roiId >> 6;
  const int chBase = split * 128;
  const int ch     = chBase + tid; // this thread's channel

  const float* roi = rois + roiId * 4;
  const float x = roi[0], y = roi[1], w = roi[2], h = roi[3];

  // Finest (n=4) bin boundaries; n=1/n=2 boundaries coincide exactly because
  // h/4, h/2 are exact power-of-two scalings (i*(h/2) == (2i)*(h/4) bit-exact).
  // jnp.round == round-half-even == rintf under default RNE mode.
  int cb[5], rb[5];
  const float h4 = h * 0.25f, w4 = w * 0.25f;
#pragma unroll
  for (int k = 0; k < 5; ++k) {
    cb[k] = (int)rintf(x + (float)k * h4);  // column (W-axis) boundaries from x,h
    rb[k] = (int)rintf(y + (float)k * w4);  // row    (H-axis) boundaries from y,w
  }

  int width = cb[4] - cb[0];
  if (width < 0) width = 0;
  if (width > MAXW) width = MAXW;
  const int chunks = width * 8;   // 16B chunks per wave per row (width cols * 128B slice)

  int c0[4], c1[4];
#pragma unroll
  for (int ix = 0; ix < 4; ++ix) {
    int a = cb[ix]   - cb[0]; if (a < 0) a = 0; if (a > width) a = width;
    int e = cb[ix+1] - cb[0]; if (e < 0) e = 0; if (e > width) e = width;
    c0[ix] = a; c1[ix] = e;
  }

  const int r0    = rb[0];
  const int nrows = rb[4] - rb[0];

  // Global base for this wave's 32-channel slice.
  const float* imgB = img + (size_t)b * (IMG_H * IMG_W * IMG_C) + chBase + wave * 32;

  // Exactly 6 async b128 issues per wave per call (pads with duplicate chunk 0,
  // identical bytes -> ASYNCcnt bookkeeping stays a compile-time constant).
  auto prefetch_row = [&](int row, int slot) {
    const float* gb = imgB + (size_t)(row * IMG_W + cb[0]) * IMG_C;
    float* db = &buf[wave][slot][0];
#pragma unroll
    for (int it = 0; it < 6; ++it) {
      int k  = lane + it * 32;
      int kk = (k < chunks) ? k : 0;
      int j  = kk >> 3;        // column within segment
      int s  = kk & 7;         // 16B chunk within this wave's 128B channel slice
      async_copy16(gb + j * IMG_C + s * 4, db + j * 32 + s * 4);
    }
  };

  float m[4][4];  // m[jy][ix] : finest 4x4 bin maxima for this thread's channel
#pragma unroll
  for (int a = 0; a < 4; ++a)
#pragma unroll
    for (int e = 0; e < 4; ++e) m[a][e] = -__builtin_inff();

  if (nrows > 0 && chunks > 0) {
    const int last = nrows - 1;
    // Prime the pipeline: rows 0,1,2 (clamped) into slots 0,1,2 -> 18 in flight.
#pragma unroll
    for (int p = 0; p < 3; ++p) {
      int rr = (p < last) ? p : last;
      prefetch_row(r0 + rr, p);
    }
    int i = 0;
#pragma unroll
    for (int jy = 0; jy < 4; ++jy) {          // unrolled -> m[][] stays in registers
      for (int row = rb[jy]; row < rb[jy + 1]; ++row, ++i) {
        int np = (i + 3 < last) ? (i + 3) : last;   // tail pads with last row (dummy)
        prefetch_row(r0 + np, (i + 3) & 3);
        wait_async<18>();                      // newest 18 = rows i+1..i+3; row i done
        const float* cur = &buf[wave][i & 3][0];
#pragma unroll
        for (int ix = 0; ix < 4; ++ix) {
          float mm = m[jy][ix];
          for (int j = c0[ix]; j < c1[ix]; ++j)
            mm = fmaxf(mm, cur[j * 32 + lane]);   // conflict-free: lane-stride 1
          m[jy][ix] = mm;
        }
      }
    }
  }

  // Emit 21 bins, bin-major then channel: [roiId][bin][ch].
  const size_t obase = (size_t)roiId * (21 * 256) + ch;

  // n = 4 level: bin 5 + ix*4 + jy (reference reshape is ix-major).
#pragma unroll
  for (int ix = 0; ix < 4; ++ix)
#pragma unroll
    for (int jy = 0; jy < 4; ++jy)
      out[obase + (size_t)(5 + ix * 4 + jy) * 256] = m[jy][ix];

  // n = 2 level: exact 2x2 reduction of finest grid.
#pragma unroll
  for (int ix2 = 0; ix2 < 2; ++ix2)
#pragma unroll
    for (int jy2 = 0; jy2 < 2; ++jy2) {
      float v = fmaxf(fmaxf(m[2*jy2][2*ix2],     m[2*jy2][2*ix2 + 1]),
                      fmaxf(m[2*jy2 + 1][2*ix2], m[2*jy2 + 1][2*ix2 + 1]));
      out[obase + (size_t)(1 + ix2 * 2 + jy2) * 256] = v;
    }

  // n = 1 level: reduction of everything.
  float g0 = fmaxf(fmaxf(m[0][0], m[0][1]), fmaxf(m[0][2], m[0][3]));
  float g1 = fmaxf(fmaxf(m[1][0], m[1][1]), fmaxf(m[1][2], m[1][3]));
  float g2 = fmaxf(fmaxf(m[2][0], m[2][1]), fmaxf(m[2][2], m[2][3]));
  float g3 = fmaxf(fmaxf(m[3][0], m[3][1]), fmaxf(m[3][2], m[3][3]));
  out[obase] = fmaxf(fmaxf(g0, g1), fmaxf(g2, g3));
}

extern "C" void kernel_launch(void* const* d_in, const int* in_sizes, int n_in,
                              void* d_out, int out_size, void* d_ws, size_t ws_size,
                              hipStream_t stream) {
  (void)in_sizes; (void)n_in; (void)d_ws; (void)ws_size; (void)out_size;
  const float* img  = (const float*)d_in[0];   // [4,32,32,256] f32
  const float* rois = (const float*)d_in[1];   // [4,64,4]      f32
  float* out = (float*)d_out;                  // [4,64,21*256] f32
  roi_pool_gfx1250<<<dim3(4 * N_ROI * 2), dim3(128), 0, stream>>>(img, rois, out);
}